// SingleColumnModel_28613072126476
// MI455X (gfx1250) — compile-verified
//
#include <hip/hip_runtime.h>

#define NCOLS 32768
#define NZ    64
#define NT    40
#define CPB   64   // columns per block == threads per block (2 wave32 waves)

#ifndef __has_builtin
#define __has_builtin(x) 0
#endif

#if defined(__AMDGCN__) && __has_builtin(__builtin_amdgcn_tensor_load_to_lds)
#define USE_TDM 1
#else
#define USE_TDM 0
#endif

#if USE_TDM
typedef unsigned int u32x4 __attribute__((ext_vector_type(4)));
typedef int i32x4 __attribute__((ext_vector_type(4)));
typedef int i32x8 __attribute__((ext_vector_type(8)));

// DMA a CPB x NZ fp32 tile (row stride NZ floats) from global into LDS,
// inserting 1 pad dword every 64 dwords -> LDS column stride 65 dwords,
// so lane L reading lds[L*65 + i] is bank-conflict free for every i.
__device__ __forceinline__ void tdm_load_tile(const float* gsrc, unsigned lds_byte_off)
{
  unsigned long long ga = (unsigned long long)gsrc;
  u32x4 g0;
  g0[0] = 1u;                                  // count=1 (valid user descriptor)
  g0[1] = lds_byte_off;                        // lds_addr (bytes)
  g0[2] = (unsigned)(ga & 0xFFFFFFFFull);      // global_addr[31:0]
  g0[3] = ((unsigned)(ga >> 32) & 0x1FFFFFFu)  // global_addr[56:32]
        | (2u << 30);                          // type = 2 ("image")
  i32x8 g1;
  g1[0] = (int)((2u << 16)                     // data_size = 4 bytes
              | (1u << 20)                     // pad_enable
              | (5u << 22));                   // pad_interval: every 64 dwords (pad_amount=0 => 1 dword)
  g1[1] = (int)(((unsigned)NZ) << 16);                  // tensor_dim0 = NZ
  g1[2] = (int)(((unsigned)NCOLS & 0xFFFFu) << 16);     // tensor_dim1[15:0]
  g1[3] = (int)((((unsigned)NCOLS >> 16) & 0xFFFFu)     // tensor_dim1[31:16]
              | (((unsigned)NZ) << 16));                // tile_dim0 = NZ
  g1[4] = (int)(CPB);                                   // tile_dim1 = CPB, tile_dim2 = 0
  g1[5] = (int)NZ;                                      // tensor_dim0_stride = NZ
  g1[6] = 0;
  g1[7] = 0;
  i32x4 z4; z4[0] = z4[1] = z4[2] = z4[3] = 0;
#if __clang_major__ >= 23
  i32x8 z8;
  #pragma unroll
  for (int k = 0; k < 8; ++k) z8[k] = 0;
  __builtin_amdgcn_tensor_load_to_lds(g0, g1, z4, z4, z8, 0);
#else
  __builtin_amdgcn_tensor_load_to_lds(g0, g1, z4, z4, 0);
#endif
}
#endif // USE_TDM

// ---------------------------------------------------------------------------
// Kernel 1: T and S diffusion (akt matrix). Factorization done once; the two
// recurrences are interleaved so the serial FMA chain is half-hidden.
// LDS [z][col] arrays: cp, e = a*invden, gh = invden*hz   (48 KB)
// Registers: t, s, dpt, dps, qgt, qgs  (~384 VGPR of live state)
// ---------------------------------------------------------------------------
__global__ __launch_bounds__(CPB) void scm_ts_kernel(
    const float* __restrict__ t0g, const float* __restrict__ s0g,
    const float* __restrict__ hzg, const float* __restrict__ akg,
    const float* __restrict__ tfg, const float* __restrict__ sfg,
    float* __restrict__ tout, float* __restrict__ sout)
{
  __shared__ float lds[3 * NZ * CPB];      // 49152 bytes
  const int c = threadIdx.x;
  const int col = blockIdx.x * CPB + c;
  const size_t cbase = (size_t)col * NZ;
  const float dtf = 30.0f;

  float t[NZ], s[NZ];

#if USE_TDM
  {
    int wid = __builtin_amdgcn_readfirstlane((int)(threadIdx.x >> 5));
    if (wid == 0) {
      unsigned lbase = (unsigned)(unsigned long long)&lds[0];
      tdm_load_tile(t0g + (size_t)blockIdx.x * CPB * NZ, lbase);
      tdm_load_tile(s0g + (size_t)blockIdx.x * CPB * NZ,
                    lbase + (unsigned)(CPB * (NZ + 1) * 4));
      __builtin_amdgcn_s_wait_tensorcnt(0);
    }
    __syncthreads();
    #pragma unroll
    for (int i = 0; i < NZ; ++i) {
      t[i] = lds[c * (NZ + 1) + i];
      s[i] = lds[CPB * (NZ + 1) + c * (NZ + 1) + i];
    }
    __syncthreads();   // LDS is about to be overwritten with coefficients
  }
#else
  #pragma unroll
  for (int i = 0; i < NZ; ++i) { t[i] = t0g[cbase + i]; s[i] = s0g[cbase + i]; }
  __syncthreads();
#endif

  // --- one-time Thomas factorization for the (time-invariant) akt matrix ---
  float qgt[NZ], qgs[NZ];
  {
    float hzr[NZ];
    #pragma unroll
    for (int i = 0; i < NZ; ++i) hzr[i] = hzg[cbase + i];
    float akr[NZ + 1];
    #pragma unroll
    for (int i = 0; i <= NZ; ++i) akr[i] = akg[(size_t)col * (NZ + 1) + i];

    float cp_prev = 0.f, g_cur = 0.f;   // g_i = -2*dt*ak[i]/(hz[i-1]+hz[i]); a_i=g_i, c_i=g_{i+1}
    #pragma unroll
    for (int i = 0; i < NZ; ++i) {
      float g_next = (i < NZ - 1) ? (-2.f * dtf * akr[i + 1]) / (hzr[i] + hzr[i + 1]) : 0.f;
      float ai = g_cur, ci = g_next;
      float bi = hzr[i] - ai - ci;
      float invd = 1.f / (bi - ai * cp_prev);
      float cp = ci * invd;
      lds[0 * NZ * CPB + i * CPB + c] = cp;              // cp
      lds[1 * NZ * CPB + i * CPB + c] = ai * invd;       // e
      lds[2 * NZ * CPB + i * CPB + c] = invd * hzr[i];   // gh
      qgt[i] = invd * dtf * tfg[cbase + i];
      qgs[i] = invd * dtf * sfg[cbase + i];
      cp_prev = cp; g_cur = g_next;
    }
  }

  // --- 40 timesteps: dp_i = gh*x_i + qg_i - e*dp_{i-1}; x_i = dp_i - cp*x_{i+1}
  #pragma unroll 1
  for (int step = 0; step < NT; ++step) {
    float dpt[NZ], dps[NZ];
    float dtp = 0.f, dsp = 0.f;
    #pragma unroll
    for (int i = 0; i < NZ; ++i) {
      float gh = lds[2 * NZ * CPB + i * CPB + c];
      float e  = lds[1 * NZ * CPB + i * CPB + c];
      dtp = gh * t[i] + qgt[i] - e * dtp; dpt[i] = dtp;
      dsp = gh * s[i] + qgs[i] - e * dsp; dps[i] = dsp;
    }
    float tn = 0.f, sn = 0.f;
    #pragma unroll
    for (int i = NZ - 1; i >= 0; --i) {
      float cp = lds[0 * NZ * CPB + i * CPB + c];
      tn = dpt[i] - cp * tn; t[i] = tn;
      sn = dps[i] - cp * sn; s[i] = sn;
    }
  }

  float4* to = (float4*)(tout + cbase);
  float4* so = (float4*)(sout + cbase);
  #pragma unroll
  for (int i = 0; i < NZ / 4; ++i) {
    to[i] = make_float4(t[4 * i], t[4 * i + 1], t[4 * i + 2], t[4 * i + 3]);
    so[i] = make_float4(s[4 * i], s[4 * i + 1], s[4 * i + 2], s[4 * i + 3]);
  }
}

// ---------------------------------------------------------------------------
// Kernel 2: U and V (akv matrix + semi-implicit Coriolis folded into the rhs).
// rhs: fu = cff1*hz*(coef*u + dtfc*v) (+surface/bottom stress at the ends);
// with invden folded:  Ag = invd*cff1*coef*hz,  Bg = invd*cff1*dtfc*hz.
// LDS [z][col]: e, Ag, Bg.  Registers: u, v, dpu, dpv, cp.
// ---------------------------------------------------------------------------
__global__ __launch_bounds__(CPB) void scm_uv_kernel(
    const float* __restrict__ u0g, const float* __restrict__ v0g,
    const float* __restrict__ hzg, const float* __restrict__ akg,
    const float* __restrict__ usfc, const float* __restrict__ vsfc,
    const float* __restrict__ ubtm, const float* __restrict__ vbtm,
    const float* __restrict__ fcg,
    float* __restrict__ uout, float* __restrict__ vout)
{
  __shared__ float lds[3 * NZ * CPB];
  const int c = threadIdx.x;
  const int col = blockIdx.x * CPB + c;
  const size_t cbase = (size_t)col * NZ;
  const float dtf = 30.0f;
  const float gam = 0.55f;

  float u[NZ], v[NZ];

#if USE_TDM
  {
    int wid = __builtin_amdgcn_readfirstlane((int)(threadIdx.x >> 5));
    if (wid == 0) {
      unsigned lbase = (unsigned)(unsigned long long)&lds[0];
      tdm_load_tile(u0g + (size_t)blockIdx.x * CPB * NZ, lbase);
      tdm_load_tile(v0g + (size_t)blockIdx.x * CPB * NZ,
                    lbase + (unsigned)(CPB * (NZ + 1) * 4));
      __builtin_amdgcn_s_wait_tensorcnt(0);
    }
    __syncthreads();
    #pragma unroll
    for (int i = 0; i < NZ; ++i) {
      u[i] = lds[c * (NZ + 1) + i];
      v[i] = lds[CPB * (NZ + 1) + c * (NZ + 1) + i];
    }
    __syncthreads();
  }
#else
  #pragma unroll
  for (int i = 0; i < NZ; ++i) { u[i] = u0g[cbase + i]; v[i] = v0g[cbase + i]; }
  __syncthreads();
#endif

  // per-column Coriolis scalars
  const float fc    = fcg[col];
  const float dtfc  = dtf * fc;
  const float cff   = dtfc * dtfc;
  const float cff1  = 1.f / (1.f + gam * gam * cff);
  const float coefc = 1.f - gam * (1.f - gam) * cff;

  float cpv[NZ];
  float bu0 = 0.f, bu1 = 0.f, bv0 = 0.f, bv1 = 0.f;
  {
    float hzr[NZ];
    #pragma unroll
    for (int i = 0; i < NZ; ++i) hzr[i] = hzg[cbase + i];
    float akr[NZ + 1];
    #pragma unroll
    for (int i = 0; i <= NZ; ++i) akr[i] = akg[(size_t)col * (NZ + 1) + i];

    float cp_prev = 0.f, g_cur = 0.f;
    #pragma unroll
    for (int i = 0; i < NZ; ++i) {
      float g_next = (i < NZ - 1) ? (-2.f * dtf * akr[i + 1]) / (hzr[i] + hzr[i + 1]) : 0.f;
      float ai = g_cur, ci = g_next;
      float bi = hzr[i] - ai - ci;
      float invd = 1.f / (bi - ai * cp_prev);
      float cp = ci * invd;
      cpv[i] = cp;
      lds[0 * NZ * CPB + i * CPB + c] = ai * invd;              // e
      float w = invd * cff1 * hzr[i];
      lds[1 * NZ * CPB + i * CPB + c] = w * coefc;              // Ag
      lds[2 * NZ * CPB + i * CPB + c] = w * dtfc;               // Bg
      if (i == 0)      { bu0 = -invd * dtf * ubtm[col]; bv0 = -invd * dtf * vbtm[col]; }
      if (i == NZ - 1) { bu1 =  invd * dtf * usfc[col]; bv1 =  invd * dtf * vsfc[col]; }
      cp_prev = cp; g_cur = g_next;
    }
  }

  #pragma unroll 1
  for (int step = 0; step < NT; ++step) {
    float dpu[NZ], dpv[NZ];
    float du = 0.f, dv = 0.f;
    #pragma unroll
    for (int i = 0; i < NZ; ++i) {
      float e  = lds[0 * NZ * CPB + i * CPB + c];
      float ag = lds[1 * NZ * CPB + i * CPB + c];
      float bg = lds[2 * NZ * CPB + i * CPB + c];
      float fu = ag * u[i] + bg * v[i];
      float fv = ag * v[i] - bg * u[i];
      if (i == 0)      { fu += bu0; fv += bv0; }
      if (i == NZ - 1) { fu += bu1; fv += bv1; }
      du = fu - e * du; dpu[i] = du;
      dv = fv - e * dv; dpv[i] = dv;
    }
    float un = 0.f, vn = 0.f;
    #pragma unroll
    for (int i = NZ - 1; i >= 0; --i) {
      float cp = cpv[i];
      un = dpu[i] - cp * un; u[i] = un;
      vn = dpv[i] - cp * vn; v[i] = vn;
    }
  }

  float4* uo = (float4*)(uout + cbase);
  float4* vo = (float4*)(vout + cbase);
  #pragma unroll
  for (int i = 0; i < NZ / 4; ++i) {
    uo[i] = make_float4(u[4 * i], u[4 * i + 1], u[4 * i + 2], u[4 * i + 3]);
    vo[i] = make_float4(v[4 * i], v[4 * i + 1], v[4 * i + 2], v[4 * i + 3]);
  }
}

extern "C" void kernel_launch(void* const* d_in, const int* in_sizes, int n_in,
                              void* d_out, int out_size, void* d_ws, size_t ws_size,
                              hipStream_t stream)
{
  (void)in_sizes; (void)n_in; (void)out_size; (void)d_ws; (void)ws_size;
  const float* u0   = (const float*)d_in[0];
  const float* v0   = (const float*)d_in[1];
  const float* t0   = (const float*)d_in[2];
  const float* s0   = (const float*)d_in[3];
  const float* hz   = (const float*)d_in[4];
  const float* akv  = (const float*)d_in[5];
  const float* akt  = (const float*)d_in[6];
  const float* tfrc = (const float*)d_in[7];
  const float* sfrc = (const float*)d_in[8];
  const float* usfc = (const float*)d_in[9];
  const float* vsfc = (const float*)d_in[10];
  const float* ubtm = (const float*)d_in[11];
  const float* vbtm = (const float*)d_in[12];
  const float* fcor = (const float*)d_in[13];

  float* out   = (float*)d_out;                 // (4, NCOLS, NZ): u, v, t, s
  float* u_out = out;
  float* v_out = out + (size_t)NCOLS * NZ;
  float* t_out = out + 2 * (size_t)NCOLS * NZ;
  float* s_out = out + 3 * (size_t)NCOLS * NZ;

  dim3 grid(NCOLS / CPB);
  dim3 block(CPB);
  scm_ts_kernel<<<grid, block, 0, stream>>>(t0, s0, hz, akt, tfrc, sfrc, t_out, s_out);
  scm_uv_kernel<<<grid, block, 0, stream>>>(u0, v0, hz, akv, usfc, vsfc, ubtm, vbtm,
                                            fcor, u_out, v_out);
}